// Focal_loss2_44040594653641
// MI455X (gfx1250) — compile-verified
//
#include <hip/hip_runtime.h>
#include <math.h>

typedef float v2f __attribute__((ext_vector_type(2)));
typedef float v8f __attribute__((ext_vector_type(8)));

__device__ __forceinline__ float sigmoidf_(float x) { return 1.0f / (1.0f + expf(-x)); }
// softplus(x) = log(1+exp(x)) = -log_sigmoid(-x), numerically stable
__device__ __forceinline__ float softplusf_(float x) {
    return fmaxf(x, 0.0f) + log1pf(expf(-fabsf(x)));
}

// ---------------------------------------------------------------------------
// Negative branch: per (b,a,z,ystrip) block.
// Stage sigmoid(logits) for rows [y0-1, y0+SY] of planes z-1,z,z+1 into LDS
// (-inf padding outside the volume == reference's reduce_window -inf init),
// then each element checks if it is the 3x3x3 max and accumulates
// sum(w) and sum(softplus(l)*w), w = s^2 * [prob_gt == -1] * [is max].
// ---------------------------------------------------------------------------
template <int D, int LOG2D, int SY>
__global__ void neg_stencil_kernel(const float* __restrict__ logits,
                                   const float* __restrict__ prob,
                                   float* __restrict__ ws_out) {
    extern __shared__ float smem[];
    constexpr int PLANE  = D * D;
    constexpr int NSTRIP = D / SY;
    constexpr int R      = SY + 2;          // rows staged per plane
    constexpr int C4     = R * D / 4;       // float4 count per staged plane

    const int tid   = threadIdx.x;          // 256 threads
    const int strip = blockIdx.x % NSTRIP;
    const int z     = (blockIdx.x / NSTRIP) % D;
    const int ba    = blockIdx.x / (NSTRIP * D);
    const int y0    = strip * SY;
    const long base = (long)ba * (long)D * PLANE;

    // CDNA5 prefetch hint for the z+2 plane of this strip (speculative, safe).
    if (z + 2 < D) {
        const int pf = tid << 4; // 16 floats (one 64B line) per thread
        if (pf < SY * D)
            __builtin_prefetch(logits + base + (long)(z + 2) * PLANE + (long)y0 * D + pf, 0, 0);
    }

    // Stage 3 planes of sigmoid values (rows y0-1 .. y0+SY), -inf outside.
    for (int dz = 0; dz < 3; ++dz) {
        const int zz = z + dz - 1;
        float4* dst = (float4*)(smem + dz * R * D);
        for (int i = tid; i < C4; i += 256) {
            const int r  = i / (D / 4);
            const int xq = i % (D / 4);
            const int gy = y0 - 1 + r;
            float4 res;
            if (zz >= 0 && zz < D && gy >= 0 && gy < D) {
                const float4 v =
                    ((const float4*)(logits + base + (long)zz * PLANE + (long)gy * D))[xq];
                res.x = sigmoidf_(v.x); res.y = sigmoidf_(v.y);
                res.z = sigmoidf_(v.z); res.w = sigmoidf_(v.w);
            } else {
                res.x = res.y = res.z = res.w = -__builtin_inff();
            }
            dst[i] = res;
        }
    }
    __syncthreads();

    const float* lg = logits + base + (long)z * PLANE + (long)y0 * D;
    const float* pg = prob   + base + (long)z * PLANE + (long)y0 * D;

    float sum_w = 0.0f, sum_nw = 0.0f;
    constexpr int NEL = SY * D;
    for (int idx = tid; idx < NEL; idx += 256) {
        const int yl = idx >> LOG2D;          // local row in strip
        const int x  = idx & (D - 1);
        const float sc = smem[(1 * R + (yl + 1)) * D + x];  // center (plane z)
        float m = sc;
        const int x0 = (x > 0) ? (x - 1) : x;
        const int x1 = (x < D - 1) ? (x + 1) : x;
#pragma unroll
        for (int dz = 0; dz < 3; ++dz) {
#pragma unroll
            for (int ry = 0; ry < 3; ++ry) {
                const float* row = smem + (dz * R + (yl + ry)) * D;
                for (int xx = x0; xx <= x1; ++xx) m = fmaxf(m, row[xx]);
            }
        }
        const float pgv = pg[idx];
        if (pgv == -1.0f && m == sc) {        // negmask * max_filter
            const float w = sc * sc;          // prob^ALPHA, ALPHA=2
            sum_w  += w;
            sum_nw += softplusf_(lg[idx]) * w;
        }
    }

    // Block reduction (deterministic tree), reuse LDS.
    __syncthreads();
    smem[tid]       = sum_w;
    smem[256 + tid] = sum_nw;
    __syncthreads();
    for (int s = 128; s > 0; s >>= 1) {
        if (tid < s) {
            smem[tid]       += smem[tid + s];
            smem[256 + tid] += smem[256 + tid + s];
        }
        __syncthreads();
    }
    if (tid == 0) {
        ws_out[2 * blockIdx.x + 0] = smem[0];
        ws_out[2 * blockIdx.x + 1] = smem[256];
    }
}

// ---------------------------------------------------------------------------
// Positive branch + tag segment-min. 1 block, 256 threads:
// thread = (level<<7) | (b<<6) | m   over levels {a,b}, B=2, M=64.
// Writes out[0], out[2], out[4] and out[6..37] (pred_prob_min [2,2,8]).
// ---------------------------------------------------------------------------
__global__ void pos_gather_kernel(const float* __restrict__ logits_a,
                                  const float* __restrict__ logits_b,
                                  const int* __restrict__ connects_a,
                                  const int* __restrict__ connects_b,
                                  const int* __restrict__ coord_a,
                                  const int* __restrict__ coord_b,
                                  float* __restrict__ out) {
    __shared__ unsigned segmin[32];   // [level][b][tag] as ordered float bits
    __shared__ float red_l[256], red_c[256];

    const int t = threadIdx.x;
    if (t < 32) segmin[t] = 0x7F800000u;  // +inf
    __syncthreads();

    const int level = t >> 7;
    const int b     = (t >> 6) & 1;
    const int m     = t & 63;
    const int D     = level ? 64 : 128;
    const float* logits  = level ? logits_b : logits_a;
    const int* connects  = level ? connects_b : connects_a;
    const int* coord     = level ? coord_b : coord_a;

    const int ci = (((b << 6) + m) << 2);
    int c0 = coord[ci + 0], c1 = coord[ci + 1], c2 = coord[ci + 2], c3 = coord[ci + 3];
    const bool valid = (c0 > -1);
    if (!valid) { c0 = 0; c1 = 0; c2 = 0; c3 = 0; }

    const long off = (((long)(b * 2 + c0) * D + c1) * D + c2) * (long)D + c3;
    const float lp  = logits[off];
    const float pp  = sigmoidf_(lp);
    const float omp = 1.0f - pp;
    const float w1  = omp * omp;                  // (1-p)^ALPHA
    const float vf  = valid ? 1.0f : 0.0f;
    // w2 = ANCHOR_POS_FACTOR[c0] == 1.0 for both anchors
    red_l[t] = softplusf_(-lp) * w1 * vf;         // -log_sigmoid(lp)*w1*w2*vf
    red_c[t] = w1 * vf;

    const int tag = connects[off] & 7;            // tags in [0, 8)
    atomicMin(&segmin[(level << 4) + (b << 3) + tag],
              __float_as_uint(valid ? w1 : __builtin_inff()));
    __syncthreads();

    // Per-level deterministic tree reduce (halves stay separate).
    for (int s = 64; s > 0; s >>= 1) {
        if ((t & 127) < s) {
            red_l[t] += red_l[t + s];
            red_c[t] += red_c[t + s];
        }
        __syncthreads();
    }
    if (t == 0) {
        const float la = red_l[0],   lb = red_l[128];
        const float ca = red_c[0],   cb = red_c[128];
        out[0] = la * 2.0f + lb * 1.0f;   // cls_loss_pos  (POS_FACTOR = [2,1])
        out[2] = ca + cb;                  // count_pos
        out[4] = ca * 2.0f + cb * 1.0f;   // wsum_pos (w2==1)
    }
    if (t < 32) {
        const float v = __uint_as_float(segmin[t]);
        out[6 + t] = (v == __builtin_inff()) ? -1.0f : v;   // absent tag -> -1
    }
}

// ---------------------------------------------------------------------------
// Final combine of negative partials, one wave (32 lanes).
// Cross-lane total via V_WMMA_F32_16X16X4_F32 ones-trick:
// A = all ones (16x4), B = {v, 0} per lane => D[m][n] = sum_k B[k][n],
// so sum_{n=0..15} D[0][n] == total of all 64 B slots (layout independent).
// ---------------------------------------------------------------------------
__device__ __forceinline__ float wave_total_wmma(float v) {
    v2f a;  a[0] = 1.0f; a[1] = 1.0f;
    v2f bm; bm[0] = v;   bm[1] = 0.0f;
    v8f c = {};
    v8f d = __builtin_amdgcn_wmma_f32_16x16x4_f32(
        /*neg_a=*/false, a, /*neg_b=*/false, bm,
        /*c_mod=*/(short)0, c, /*reuse_a=*/false, /*reuse_b=*/false);
    float t = d[0];                 // lanes 0..15 hold D[0][N=lane]
    t += __shfl_xor(t, 1);
    t += __shfl_xor(t, 2);
    t += __shfl_xor(t, 4);
    t += __shfl_xor(t, 8);          // lane 0 now has sum over N=0..15
    return t;
}

__global__ void neg_combine_kernel(const float* __restrict__ ws,
                                   float* __restrict__ out,
                                   int nA, int nB, int offB) {
    const int lane = threadIdx.x;   // 32 threads == one wave, EXEC all ones
    float swA = 0.f, snwA = 0.f, swB = 0.f, snwB = 0.f;
    for (int j = lane; j < nA; j += 32) { swA += ws[2 * j];        snwA += ws[2 * j + 1]; }
    for (int j = lane; j < nB; j += 32) { swB += ws[offB + 2 * j]; snwB += ws[offB + 2 * j + 1]; }

    const float tswA = wave_total_wmma(swA);
    const float tnwA = wave_total_wmma(snwA);
    const float tswB = wave_total_wmma(swB);
    const float tnwB = wave_total_wmma(snwB);

    if (lane == 0) {
        out[1] = tnwA * 2.0f + tnwB * 1.0f;   // cls_loss_neg (NEG_FACTOR = [2,1])
        out[3] = tswA + tswB;                  // count_neg
        out[5] = tswA * 2.0f + tswB * 1.0f;   // wsum_neg
    }
}

// ---------------------------------------------------------------------------
extern "C" void kernel_launch(void* const* d_in, const int* in_sizes, int n_in,
                              void* d_out, int out_size, void* d_ws, size_t ws_size,
                              hipStream_t stream) {
    const float* logits_a   = (const float*)d_in[0];
    const float* logits_b   = (const float*)d_in[1];
    const float* prob_a     = (const float*)d_in[2];
    const float* prob_b     = (const float*)d_in[3];
    const int*   connects_a = (const int*)d_in[4];
    const int*   connects_b = (const int*)d_in[5];
    const int*   coord_a    = (const int*)d_in[6];
    const int*   coord_b    = (const int*)d_in[7];
    float* out = (float*)d_out;
    float* ws  = (float*)d_ws;

    constexpr int BA = 4;              // B*A = 2*2
    constexpr int Da = 128, Db = 64;
    constexpr int SY = 32;
    const int nBlkA = BA * Da * (Da / SY);   // 2048
    const int nBlkB = BA * Db * (Db / SY);   // 512
    const size_t shA = (size_t)3 * (SY + 2) * Da * sizeof(float);  // 52 KB
    const size_t shB = (size_t)3 * (SY + 2) * Db * sizeof(float);  // 26 KB

    neg_stencil_kernel<Da, 7, SY><<<nBlkA, 256, shA, stream>>>(logits_a, prob_a, ws);
    neg_stencil_kernel<Db, 6, SY><<<nBlkB, 256, shB, stream>>>(logits_b, prob_b, ws + 2 * nBlkA);
    pos_gather_kernel<<<1, 256, 0, stream>>>(logits_a, logits_b, connects_a, connects_b,
                                             coord_a, coord_b, out);
    neg_combine_kernel<<<1, 32, 0, stream>>>(ws, out, nBlkA, nBlkB, 2 * nBlkA);
}